// GATModel_84799834292766
// MI455X (gfx1250) — compile-verified
//
#include <hip/hip_runtime.h>
#include <math.h>

// GAT layer for MI455X (gfx1250, wave32).
// Stage 1 uses V_WMMA_F32_16X16X4_F32 (fp32 matrix cores) for the big
// 50000x256 @ 256x256 projection GEMM. Each wave computes a 32x64 output
// block (2 M-tiles x 4 N-tiles) so A/B fragments are reused in registers:
// 8 WMMAs per 10 loads per K-step. Edge stages are bandwidth-bound
// gather/scatter; they use coalesced wave-per-edge access + atomics.

#define IN_F   256
#define OUT_F  64
#define HEADS  4
#define COLS   (HEADS * OUT_F)   // 256 output columns (h*64+o)
#define NEG_SLOPE 0.2f

typedef __attribute__((ext_vector_type(2))) float v2f;
typedef __attribute__((ext_vector_type(8))) float v8f;

// ---- order-preserving float<->uint encoding for atomicMax on floats ----
__device__ __forceinline__ unsigned fenc(float f) {
    unsigned u = __float_as_uint(f);
    return (u & 0x80000000u) ? ~u : (u | 0x80000000u);
}
__device__ __forceinline__ float fdec(unsigned u) {
    unsigned v = (u & 0x80000000u) ? (u & 0x7FFFFFFFu) : ~u;
    return __uint_as_float(v);
}
__device__ __forceinline__ float leaky(float t) {
    return t > 0.f ? t : NEG_SLOPE * t;
}

// ---- P0: zero the reduction buffers (workspace is poisoned by harness) ----
__global__ void gat_init(unsigned* __restrict__ menc, float* __restrict__ ssum,
                         float* __restrict__ agg, int nh, int nagg) {
    int i = blockIdx.x * blockDim.x + threadIdx.x;
    int stride = gridDim.x * blockDim.x;
    for (int k = i; k < nh; k += stride) { menc[k] = 0u; ssum[k] = 0.f; }
    for (int k = i; k < nagg; k += stride) agg[k] = 0.f;
}

// ---- P1: feat[n, h*64+o] = sum_i x[n,i] * W[h,i,o]  via fp32 WMMA ----
// One wave -> 32 rows x 64 cols (2 M-tiles x 4 col-tiles, one head).
// grid = (ceil(mtiles/8), HEADS), 4 waves (128 thr) per block.
__global__ void gat_feat_wmma(const float* __restrict__ x,
                              const float* __restrict__ W,
                              float* __restrict__ feat, int N) {
    const int wave = threadIdx.x >> 5;
    const int lane = threadIdx.x & 31;
    const int half = lane >> 4;      // 0: lanes 0-15, 1: lanes 16-31
    const int lrow = lane & 15;
    const int mtiles = (N + 15) >> 4;            // 3125 for N = 50000
    const int mt0 = (blockIdx.x * 4 + wave) * 2; // first of this wave's 2 M-tiles
    if (mt0 >= mtiles) return;                   // wave-uniform branch

    const int h = blockIdx.y;                    // head: 64 contiguous columns
    const float* __restrict__ Wh = W + (size_t)h * IN_F * OUT_F;

    // row indices for the two M-tiles (clamp 2nd tile; stores are guarded)
    const int r0 = (mt0 << 4) + lrow;
    const int r1 = min((mt0 << 4) + 16 + lrow, N - 1);
    const float* __restrict__ a0p = x + (size_t)r0 * IN_F;
    const float* __restrict__ a1p = x + (size_t)r1 * IN_F;

    __builtin_prefetch(a0p, 0, 1);               // global_prefetch_b8
    __builtin_prefetch(a1p, 0, 1);

    v8f acc[2][4] = {};
    for (int k = 0; k < IN_F; k += 4) {
        // A 16x4 fp32 fragments: lanes 0-15 carry K=k..k+1, lanes 16-31 K=k+2..k+3
        const int kr0 = k + 2 * half;
        const int kr1 = kr0 + 1;
        v2f a0, a1;
        a0.x = a0p[kr0]; a0.y = a0p[kr1];
        a1.x = a1p[kr0]; a1.y = a1p[kr1];
#pragma unroll
        for (int c = 0; c < 4; ++c) {
            // B 4x16 fp32 fragment for column tile c of this head
            v2f b;
            b.x = Wh[(size_t)kr0 * OUT_F + c * 16 + lrow];
            b.y = Wh[(size_t)kr1 * OUT_F + c * 16 + lrow];
            acc[0][c] = __builtin_amdgcn_wmma_f32_16x16x4_f32(
                false, a0, false, b, (short)0, acc[0][c], false, false);
            acc[1][c] = __builtin_amdgcn_wmma_f32_16x16x4_f32(
                false, a1, false, b, (short)0, acc[1][c], false, false);
        }
    }
    // D 16x16 fp32: VGPR r -> row base + r + 8*half, col = h*64 + c*16 + lrow
#pragma unroll
    for (int t = 0; t < 2; ++t) {
        const int mbase = (mt0 + t) << 4;
        if (mbase >= N) break;                   // wave-uniform
#pragma unroll
        for (int c = 0; c < 4; ++c) {
#pragma unroll
            for (int r = 0; r < 8; ++r) {
                int row = mbase + r + 8 * half;
                if (row < N)
                    feat[(size_t)row * COLS + h * OUT_F + c * 16 + lrow] = acc[t][c][r];
            }
        }
    }
}

// ---- P2: el/er[n,h] = dot(feat[n,h,:], attn_{l,r}[h,:]) ----
__global__ void gat_attn_logits(const float* __restrict__ feat,
                                const float* __restrict__ attn_l,
                                const float* __restrict__ attn_r,
                                float* __restrict__ el, float* __restrict__ er,
                                int N) {
    int i = blockIdx.x * blockDim.x + threadIdx.x;
    if (i >= N * HEADS) return;
    int h = i & (HEADS - 1);
    int n = i >> 2;
    const float* __restrict__ f  = feat   + (size_t)n * COLS + h * OUT_F;
    const float* __restrict__ al = attn_l + h * OUT_F;
    const float* __restrict__ ar = attn_r + h * OUT_F;
    float sl = 0.f, sr = 0.f;
#pragma unroll 8
    for (int o = 0; o < OUT_F; ++o) { float v = f[o]; sl += v * al[o]; sr += v * ar[o]; }
    el[i] = sl; er[i] = sr;
}

// ---- P3: segment max of leaky(el[src]+er[dst]) over dst ----
__global__ void gat_edge_max(const float* __restrict__ el, const float* __restrict__ er,
                             const int* __restrict__ src, const int* __restrict__ dst,
                             unsigned* __restrict__ menc, int E) {
    int i = blockIdx.x * blockDim.x + threadIdx.x;
    if (i >= E) return;
    int s = src[i], d = dst[i];
#pragma unroll
    for (int h = 0; h < HEADS; ++h) {
        float t = leaky(el[s * HEADS + h] + er[d * HEADS + h]);
        atomicMax(&menc[d * HEADS + h], fenc(t));
    }
}

// ---- P4: segment sum of exp(e - m[dst]) over dst ----
__global__ void gat_expsum(const float* __restrict__ el, const float* __restrict__ er,
                           const unsigned* __restrict__ menc,
                           const int* __restrict__ src, const int* __restrict__ dst,
                           float* __restrict__ ssum, int E) {
    int i = blockIdx.x * blockDim.x + threadIdx.x;
    if (i >= E * HEADS) return;
    int e = i >> 2, h = i & (HEADS - 1);
    int s = src[e], d = dst[e];
    float t = leaky(el[s * HEADS + h] + er[d * HEADS + h]);
    float m = fdec(menc[d * HEADS + h]);
    atomicAdd(&ssum[d * HEADS + h], __expf(t - m));
}

// ---- P5: agg[dst] += alpha * feat[src]  (one wave per edge, coalesced) ----
__global__ void gat_aggregate(const float* __restrict__ feat,
                              const float* __restrict__ el, const float* __restrict__ er,
                              const unsigned* __restrict__ menc, const float* __restrict__ ssum,
                              const int* __restrict__ src, const int* __restrict__ dst,
                              float* __restrict__ agg, int E) {
    int wave = threadIdx.x >> 5;
    int lane = threadIdx.x & 31;
    int e = blockIdx.x * (blockDim.x >> 5) + wave;
    if (e >= E) return;
    int s = src[e], d = dst[e];

    // every 4-lane group redundantly computes alpha for head (lane&3), then broadcast
    int hh = lane & (HEADS - 1);
    float t  = leaky(el[s * HEADS + hh] + er[d * HEADS + hh]);
    float m  = fdec(menc[d * HEADS + hh]);
    float ex = __expf(t - m);
    float ss = ssum[d * HEADS + hh];
    float a  = ex / (ss == 0.f ? 1.f : ss);

    float alpha[HEADS];
#pragma unroll
    for (int h = 0; h < HEADS; ++h) alpha[h] = __shfl(a, h, 32);

    const float* __restrict__ fr = feat + (size_t)s * COLS;
    float* __restrict__ ar = agg + (size_t)d * COLS;
#pragma unroll
    for (int h = 0; h < HEADS; ++h) {
#pragma unroll
        for (int j = 0; j < 2; ++j) {
            int o = h * OUT_F + j * 32 + lane;   // 32 contiguous lanes -> coalesced
            atomicAdd(&ar[o], fr[o] * alpha[h]);
        }
    }
}

// ---- P6: out[n,h,d] = sum_o (agg[n,h,o]+bias[h,o]) * fc_w[o,d] + fc_b[d] ----
__global__ void gat_final(const float* __restrict__ agg, const float* __restrict__ bias,
                          const float* __restrict__ fcw, const float* __restrict__ fcb,
                          float* __restrict__ out, int N) {
    int i = blockIdx.x * blockDim.x + threadIdx.x;
    if (i >= N * HEADS * 10) return;
    int dcol = i % 10;
    int nh = i / 10;
    int h = nh & (HEADS - 1);
    int n = nh >> 2;
    const float* __restrict__ a = agg  + (size_t)n * COLS + h * OUT_F;
    const float* __restrict__ b = bias + h * OUT_F;
    float acc = fcb[dcol];
#pragma unroll 8
    for (int o = 0; o < OUT_F; ++o) acc += (a[o] + b[o]) * fcw[o * 10 + dcol];
    out[i] = acc;
}

extern "C" void kernel_launch(void* const* d_in, const int* in_sizes, int n_in,
                              void* d_out, int out_size, void* d_ws, size_t ws_size,
                              hipStream_t stream) {
    const float* x      = (const float*)d_in[0];
    const int*   src    = (const int*)  d_in[1];
    const int*   dst    = (const int*)  d_in[2];
    const float* W      = (const float*)d_in[3];
    const float* attn_l = (const float*)d_in[4];
    const float* attn_r = (const float*)d_in[5];
    const float* bias   = (const float*)d_in[6];
    const float* fcw    = (const float*)d_in[7];
    const float* fcb    = (const float*)d_in[8];
    float* out = (float*)d_out;

    const int N = in_sizes[0] / IN_F;   // 50000
    const int E = in_sizes[1];          // 1200000

    // workspace layout (~104 MB)
    char* p = (char*)d_ws;
    float*    feat = (float*)p;    p += (size_t)N * COLS  * sizeof(float);
    float*    el   = (float*)p;    p += (size_t)N * HEADS * sizeof(float);
    float*    er   = (float*)p;    p += (size_t)N * HEADS * sizeof(float);
    unsigned* menc = (unsigned*)p; p += (size_t)N * HEADS * sizeof(unsigned);
    float*    ssum = (float*)p;    p += (size_t)N * HEADS * sizeof(float);
    float*    agg  = (float*)p;    p += (size_t)N * COLS  * sizeof(float);

    gat_init<<<2048, 256, 0, stream>>>(menc, ssum, agg, N * HEADS, N * COLS);

    const int mtiles = (N + 15) / 16;               // 3125
    dim3 gg((mtiles + 7) / 8, HEADS);               // 2 M-tiles per wave, 4 waves/block
    gat_feat_wmma<<<gg, 128, 0, stream>>>(x, W, feat, N);

    gat_attn_logits<<<(N * HEADS + 255) / 256, 256, 0, stream>>>(feat, attn_l, attn_r, el, er, N);
    gat_edge_max<<<(E + 255) / 256, 256, 0, stream>>>(el, er, src, dst, menc, E);
    gat_expsum<<<(E * HEADS + 255) / 256, 256, 0, stream>>>(el, er, menc, src, dst, ssum, E);
    gat_aggregate<<<(E + 7) / 8, 256, 0, stream>>>(feat, el, er, menc, ssum, src, dst, agg, E);
    gat_final<<<(N * HEADS * 10 + 255) / 256, 256, 0, stream>>>(agg, bias, fcw, fcb, out, N);
}